// VanillaCGN_438086664818
// MI455X (gfx1250) — compile-verified
//
#include <hip/hip_runtime.h>

#define NN 50000       // N_NODES
#define NE 800000      // N_EDGES
#define DD 256         // D
#define NL 3           // N_LAYERS
#define BN 4           // 16x16 N-tiles per wave (16x64 output per wave)

typedef __attribute__((ext_vector_type(2))) float v2f;
typedef __attribute__((ext_vector_type(8))) float v8f;

// ---------------------------------------------------------------- utilities
__global__ __launch_bounds__(256) void zero_kernel(float* __restrict__ p, int n) {
    int i = blockIdx.x * blockDim.x + threadIdx.x;
    if (i < n) p[i] = 0.0f;
}

__global__ __launch_bounds__(256) void deg_kernel(const long long* __restrict__ tgt,
                                                  float* __restrict__ deg, int e) {
    int i = blockIdx.x * blockDim.x + threadIdx.x;
    if (i < e) atomicAdd(deg + (int)tgt[i], 1.0f);
}

// Scatter-add: agg[tgt[e]] += h[src[e]] ; 64 threads per edge, float4 per thread.
__global__ __launch_bounds__(256) void scatter_kernel(const float* __restrict__ h,
                                                      const long long* __restrict__ src,
                                                      const long long* __restrict__ tgt,
                                                      float* __restrict__ agg, int e) {
    long long tid = (long long)blockIdx.x * blockDim.x + threadIdx.x;
    int edge = (int)(tid >> 6);
    int j    = (int)(tid & 63) << 2;
    if (edge < e) {
        int s = (int)src[edge];
        int t = (int)tgt[edge];
        const float4 v = *(const float4*)(h + (long long)s * DD + j);
        float* o = agg + (long long)t * DD + j;
        atomicAdd(o + 0, v.x);
        atomicAdd(o + 1, v.y);
        atomicAdd(o + 2, v.z);
        atomicAdd(o + 3, v.w);
    }
}

// ------------------------------------------------- WMMA f32 16x16x4 GEMMs
// One wave computes a 16(M) x 64(N) output strip: 4 accumulators, each A
// fragment reused by 4 WMMAs.
// A layout (16x4, MxK): lanes 0-15 M=lane {K=0,1}; lanes 16-31 M=lane-16 {K=2,3}
// B layout (4x16, KxN): lanes 0-15 N=lane {K=0,1}; lanes 16-31 N=lane-16 {K=2,3}
// C/D layout (16x16):   VGPR j: lanes 0-15 -> (M=j,   N=lane),
//                               lanes 16-31 -> (M=j+8, N=lane-16)

// H = X @ U0 + b0    (U0 accessed row-major: B[k][n] = U0[k*D + n])
__global__ __launch_bounds__(32) void gemm_bias_kernel(const float* __restrict__ X,
                                                       const float* __restrict__ U0,
                                                       const float* __restrict__ b0,
                                                       float* __restrict__ H) {
    const int lane = threadIdx.x;
    const int half = lane >> 4;       // 0 or 1
    const int lrow = lane & 15;
    const int m0 = blockIdx.y * 16;
    const int n_base = blockIdx.x * (16 * BN);

    v8f c[BN];
#pragma unroll
    for (int t = 0; t < BN; ++t) {
        const float bias = b0[n_base + t * 16 + lrow];
#pragma unroll
        for (int j = 0; j < 8; ++j) c[t][j] = bias;
    }

    const float* xrow = X + (long long)(m0 + lrow) * DD;
#pragma unroll 4
    for (int k = 0; k < DD; k += 4) {
        v2f a;
        a.x = xrow[k + 2 * half];
        a.y = xrow[k + 2 * half + 1];
        const float* brow0 = U0 + (long long)(k + 2 * half) * DD + lrow;
#pragma unroll
        for (int t = 0; t < BN; ++t) {
            v2f b;
            b.x = brow0[n_base + t * 16];
            b.y = brow0[n_base + t * 16 + DD];
            c[t] = __builtin_amdgcn_wmma_f32_16x16x4_f32(false, a, false, b,
                                                         (short)0, c[t], false, false);
        }
    }
#pragma unroll
    for (int t = 0; t < BN; ++t) {
        const int col = n_base + t * 16 + lrow;
#pragma unroll
        for (int j = 0; j < 8; ++j) {
            int row = m0 + j + 8 * half;
            H[(long long)row * DD + col] = c[t][j];
        }
    }
}

// OUT = relu((AGG @ U^T) / deg)   (B[k][n] = U[n*D + k] -> row-major read of U)
__global__ __launch_bounds__(32) void gemm_layer_kernel(const float* __restrict__ AGG,
                                                        const float* __restrict__ U,
                                                        const float* __restrict__ deg,
                                                        float* __restrict__ OUT) {
    const int lane = threadIdx.x;
    const int half = lane >> 4;
    const int lrow = lane & 15;
    const int m0 = blockIdx.y * 16;
    const int n_base = blockIdx.x * (16 * BN);

    v8f c[BN];
#pragma unroll
    for (int t = 0; t < BN; ++t) {
#pragma unroll
        for (int j = 0; j < 8; ++j) c[t][j] = 0.0f;
    }

    const float* arow = AGG + (long long)(m0 + lrow) * DD;
    const float* urow[BN];
#pragma unroll
    for (int t = 0; t < BN; ++t)
        urow[t] = U + (long long)(n_base + t * 16 + lrow) * DD;  // column of U^T

#pragma unroll 4
    for (int k = 0; k < DD; k += 4) {
        v2f a;
        a.x = arow[k + 2 * half];
        a.y = arow[k + 2 * half + 1];
#pragma unroll
        for (int t = 0; t < BN; ++t) {
            v2f b;
            b.x = urow[t][k + 2 * half];
            b.y = urow[t][k + 2 * half + 1];
            c[t] = __builtin_amdgcn_wmma_f32_16x16x4_f32(false, a, false, b,
                                                         (short)0, c[t], false, false);
        }
    }

    float rdeg[8];
#pragma unroll
    for (int j = 0; j < 8; ++j) rdeg[j] = 1.0f / deg[m0 + j + 8 * half];

#pragma unroll
    for (int t = 0; t < BN; ++t) {
        const int col = n_base + t * 16 + lrow;
#pragma unroll
        for (int j = 0; j < 8; ++j) {
            int row = m0 + j + 8 * half;
            OUT[(long long)row * DD + col] = fmaxf(c[t][j] * rdeg[j], 0.0f);
        }
    }
}

// ------------------------------------------------------------------- launch
extern "C" void kernel_launch(void* const* d_in, const int* in_sizes, int n_in,
                              void* d_out, int out_size, void* d_ws, size_t ws_size,
                              hipStream_t stream) {
    const float*      x    = (const float*)d_in[0];
    const long long*  edge = (const long long*)d_in[1];
    const float*      U0   = (const float*)d_in[2];
    const float*      b0   = (const float*)d_in[3];
    const float*      Us   = (const float*)d_in[4];
    float*            out  = (float*)d_out;

    const long long* src = edge;        // edge_index[0]
    const long long* tgt = edge + NE;   // edge_index[1]

    // workspace: deg [NN] | h [NN*DD] | agg [NN*DD]   (all f32, 16B aligned)
    float* deg  = (float*)d_ws;
    float* hbuf = deg + NN;
    float* agg  = hbuf + (size_t)NN * DD;

    // 1) in-degree
    zero_kernel<<<(NN + 255) / 256, 256, 0, stream>>>(deg, NN);
    deg_kernel<<<(NE + 255) / 256, 256, 0, stream>>>(tgt, deg, NE);

    // 2) h = x @ U0 + b0
    dim3 gtile(DD / (16 * BN), NN / 16);   // 4 x 3125 strips, 1 wave each
    gemm_bias_kernel<<<gtile, 32, 0, stream>>>(x, U0, b0, hbuf);

    // 3) layers
    const int nd = NN * DD;
    const long long scat_threads = (long long)NE * 64;
    const unsigned scat_blocks = (unsigned)(scat_threads / 256);
    for (int l = 0; l < NL; ++l) {
        zero_kernel<<<(nd + 255) / 256, 256, 0, stream>>>(agg, nd);
        scatter_kernel<<<scat_blocks, 256, 0, stream>>>(hbuf, src, tgt, agg, NE);
        float* dst = (l == NL - 1) ? out : hbuf;
        gemm_layer_kernel<<<gtile, 32, 0, stream>>>(agg, Us + (size_t)l * DD * DD,
                                                    deg, dst);
    }
}